// HausdorffLoss_9887014715551
// MI455X (gfx1250) — compile-verified
//
#include <hip/hip_runtime.h>
#include <math.h>

// ---------------------------------------------------------------------------
// Hausdorff (symmetric max-of-min) distance between point clouds via
// V_WMMA_F32_16X16X4_F32.  D = A(16x4) * B(4x16) + C(16x16), where
//   A row n  = (-2x0, -2x1, -2x2, |x|^2)
//   B col m  = ( y0,   y1,   y2,  1    )
//   C[n][m]  = |y_m|^2
// so D[n][m] = |x_n|^2 - 2 x_n.y_m + |y_m|^2  == squared Euclidean distance.
// sqrt is monotone -> min/max over squared distances, sqrt at the very end.
// ---------------------------------------------------------------------------

typedef float v2f __attribute__((ext_vector_type(2)));
typedef float v8f __attribute__((ext_vector_type(8)));

#define BN 8192                   // points per batch (N == M)
#define NBATCH 4
#define TOTAL (BN * NBATCH)       // 32768 points per cloud
#define ROWS_PER_WAVE 32          // two 16-row WMMA tiles per wave
#define WAVES_PER_BLOCK 8         // 256-thread block = 8 wave32

// ---- pack kernel: float3 -> float4 (x, y, z, |p|^2) --------------------------
__global__ __launch_bounds__(256) void pack_kernel(const float* __restrict__ pred,
                                                   const float* __restrict__ label,
                                                   float4* __restrict__ packP,
                                                   float4* __restrict__ packL) {
    int i = blockIdx.x * 256 + threadIdx.x;
    if (i < TOTAL) {
        float x = pred[3 * i + 0], y = pred[3 * i + 1], z = pred[3 * i + 2];
        packP[i] = make_float4(x, y, z, x * x + y * y + z * z);
        x = label[3 * i + 0]; y = label[3 * i + 1]; z = label[3 * i + 2];
        packL[i] = make_float4(x, y, z, x * x + y * y + z * z);
    }
}

__device__ __forceinline__ v8f wmma_sqdist(v2f a, v2f b, v8f c) {
    // 8 args: (neg_a, A, neg_b, B, c_mod, C, reuse_a, reuse_b)
    return __builtin_amdgcn_wmma_f32_16x16x4_f32(false, a, false, b, (short)0, c,
                                                 false, false);
}

// ---- main kernel: per-row min of squared distance over the full opposite set.
// blockIdx.y == 0 : rows = pred  points, mins over label -> minXY
// blockIdx.y == 1 : rows = label points, mins over pred  -> minYX
__global__ __launch_bounds__(256) void chamfer_min_kernel(const float4* __restrict__ packP,
                                                          const float4* __restrict__ packL,
                                                          float* __restrict__ minXY,
                                                          float* __restrict__ minYX) {
    const int lane      = threadIdx.x & 31;
    const int l15       = lane & 15;
    const int wave      = blockIdx.x * WAVES_PER_BLOCK + (threadIdx.x >> 5);
    const int wavesPerB = BN / ROWS_PER_WAVE;              // 256
    const int b         = wave / wavesPerB;
    const int rowStart  = (wave % wavesPerB) * ROWS_PER_WAVE;

    const float4* rp; const float4* cp; float* outv;
    if (blockIdx.y == 0) { rp = packP; cp = packL; outv = minXY; }
    else                 { rp = packL; cp = packP; outv = minYX; }

    // A operands for the wave's two 16-row tiles (ISA 16x4 f32 A layout:
    // lanes 0-15 hold K=0,1; lanes 16-31 hold K=2,3; row = lane&15).
    float4 p0 = rp[b * BN + rowStart + l15];
    float4 p1 = rp[b * BN + rowStart + 16 + l15];
    const bool lo = (lane < 16);
    v2f a0, a1;
    a0[0] = lo ? -2.0f * p0.x : -2.0f * p0.z;
    a0[1] = lo ? -2.0f * p0.y : p0.w;
    a1[0] = lo ? -2.0f * p1.x : -2.0f * p1.z;
    a1[1] = lo ? -2.0f * p1.y : p1.w;

    const float4* cbase = cp + b * BN;

    v8f rmin0, rmin1;
#pragma unroll
    for (int i = 0; i < 8; ++i) { rmin0[i] = 3.402823466e38f; rmin1[i] = 3.402823466e38f; }

    for (int ct = 0; ct < BN / 16; ++ct) {
        float4 q = cbase[ct * 16 + l15];         // one b128 load per lane
        v2f bb;                                  // B: K across halves, col = lane&15
        bb[0] = lo ? q.x : q.z;
        bb[1] = lo ? q.y : 1.0f;
        v8f c;                                   // C[n][m] = |y_m|^2 (column bcast)
#pragma unroll
        for (int i = 0; i < 8; ++i) c[i] = q.w;

        v8f d0 = wmma_sqdist(a0, bb, c);
        v8f d1 = wmma_sqdist(a1, bb, c);
#pragma unroll
        for (int i = 0; i < 8; ++i) {
            rmin0[i] = fminf(rmin0[i], d0[i]);
            rmin1[i] = fminf(rmin1[i], d1[i]);
        }
    }

    // Cross-lane min within each 16-lane half (columns of the tile).
    // After this: lanes 0-15 hold rows 0-7 mins, lanes 16-31 hold rows 8-15.
#pragma unroll
    for (int off = 8; off >= 1; off >>= 1) {
#pragma unroll
        for (int i = 0; i < 8; ++i) {
            rmin0[i] = fminf(rmin0[i], __shfl_xor(rmin0[i], off, 32));
            rmin1[i] = fminf(rmin1[i], __shfl_xor(rmin1[i], off, 32));
        }
    }

    if (l15 == 0) {                              // lanes 0 and 16 write out
        int base = b * BN + rowStart + ((lane >> 4) << 3);  // +0 or +8
#pragma unroll
        for (int i = 0; i < 8; ++i) {
            outv[base + i]      = fmaxf(rmin0[i], 0.0f);    // tile 0: rows +0..15
            outv[base + 16 + i] = fmaxf(rmin1[i], 0.0f);    // tile 1: rows +16..31
        }
    }
}

// ---- finalize: global max over each min-array, sqrt, sum -> scalar ----------
__global__ __launch_bounds__(256) void finalize_kernel(const float* __restrict__ minXY,
                                                       const float* __restrict__ minYX,
                                                       float* __restrict__ out) {
    __shared__ float s[256];
    const int tid = threadIdx.x;

    float mx = 0.0f;
    for (int i = tid; i < TOTAL; i += 256) mx = fmaxf(mx, minXY[i]);
    s[tid] = mx; __syncthreads();
    for (int o = 128; o > 0; o >>= 1) { if (tid < o) s[tid] = fmaxf(s[tid], s[tid + o]); __syncthreads(); }
    float dis_xy = sqrtf(s[0]);
    __syncthreads();

    float my = 0.0f;
    for (int i = tid; i < TOTAL; i += 256) my = fmaxf(my, minYX[i]);
    s[tid] = my; __syncthreads();
    for (int o = 128; o > 0; o >>= 1) { if (tid < o) s[tid] = fmaxf(s[tid], s[tid + o]); __syncthreads(); }

    if (tid == 0) out[0] = dis_xy + sqrtf(s[0]);
}

extern "C" void kernel_launch(void* const* d_in, const int* in_sizes, int n_in,
                              void* d_out, int out_size, void* d_ws, size_t ws_size,
                              hipStream_t stream) {
    const float* pred  = (const float*)d_in[0];   // [4, 8192, 3] f32
    const float* label = (const float*)d_in[1];   // [4, 8192, 3] f32
    float* out = (float*)d_out;                   // scalar f32

    char* ws = (char*)d_ws;                       // 1.25 MB used
    float4* packP = (float4*)(ws);
    float4* packL = (float4*)(ws + (size_t)TOTAL * sizeof(float4));
    float*  minXY = (float*)(ws + 2ull * TOTAL * sizeof(float4));
    float*  minYX = (float*)(ws + 2ull * TOTAL * sizeof(float4) + (size_t)TOTAL * sizeof(float));

    pack_kernel<<<(TOTAL + 255) / 256, 256, 0, stream>>>(pred, label, packP, packL);

    dim3 grid(TOTAL / (ROWS_PER_WAVE * WAVES_PER_BLOCK), 2);   // (128, 2)
    chamfer_min_kernel<<<grid, 256, 0, stream>>>(packP, packL, minXY, minYX);

    finalize_kernel<<<1, 256, 0, stream>>>(minXY, minYX, out);
}